// LAGCN_84267258347715
// MI455X (gfx1250) — compile-verified
//
#include <hip/hip_runtime.h>

typedef __attribute__((ext_vector_type(2))) float v2f;
typedef __attribute__((ext_vector_type(8))) float v8f;

#define F_IN  128
#define H_DIM 64
#define D_MID 128
#define OUTF  40
#define BN_EPS 1e-5f

// ---------------------------------------------------------------- utilities
__global__ void fill_kernel(float* __restrict__ p, float v, int n) {
    int i = blockIdx.x * blockDim.x + threadIdx.x;
    if (i < n) p[i] = v;
}

__global__ void deg_kernel(const int* __restrict__ dst, float* __restrict__ deg, int E) {
    int e = blockIdx.x * blockDim.x + threadIdx.x;
    if (e < E) atomicAdd(&deg[dst[e]], 1.0f);
}

__global__ void dinv_kernel(float* __restrict__ d, int n) {
    int i = blockIdx.x * blockDim.x + threadIdx.x;
    if (i < n) d[i] = rsqrtf(d[i]);   // deg >= 1 always (self-loop)
}

__global__ void norm_kernel(const int* __restrict__ src, const int* __restrict__ dst,
                            const float* __restrict__ dinv, float* __restrict__ norm, int E) {
    int e = blockIdx.x * blockDim.x + threadIdx.x;
    if (e < E) norm[e] = dinv[src[e]] * dinv[dst[e]];
}

// out[n,c] = bias[c] + dinv[n]^2 * h[n,c]   (self-loop term; full overwrite)
__global__ void selfloop_init_kernel(const float* __restrict__ h, const float* __restrict__ dinv,
                                     const float* __restrict__ bias, float* __restrict__ out,
                                     int total, int F) {
    int i = blockIdx.x * blockDim.x + threadIdx.x;
    if (i >= total) return;
    int n = i / F, c = i - n * F;
    float d = dinv[n];
    float b = bias ? bias[c] : 0.0f;
    out[i] = b + d * d * h[i];
}

// edge scatter: out[dst] += norm[e] * h[src], float4-vectorized per edge
__global__ void agg_kernel(const int* __restrict__ src, const int* __restrict__ dst,
                           const float* __restrict__ norm, const float* __restrict__ h,
                           float* __restrict__ out, int E, int F4, int F) {
    int gid = blockIdx.x * blockDim.x + threadIdx.x;
    if (gid >= E * F4) return;
    int e = gid / F4;
    int q = (gid - e * F4) << 2;
    int s = src[e], d = dst[e];
    float w = norm[e];
    const float4 v = *reinterpret_cast<const float4*>(h + (size_t)s * F + q);
    float* o = out + (size_t)d * F + q;
    atomicAdd(o + 0, v.x * w);
    atomicAdd(o + 1, v.y * w);
    atomicAdd(o + 2, v.z * w);
    atomicAdd(o + 3, v.w * w);
}

// per-column mean / rsqrt(var+eps): one block (256 thr) per column
__global__ void bn_stats_kernel(const float* __restrict__ x, float* __restrict__ mu,
                                float* __restrict__ rsig, int N, int F) {
    __shared__ float s_sum[256], s_sq[256];
    int c = blockIdx.x, t = threadIdx.x;
    float sum = 0.0f, sq = 0.0f;
    for (int r = t; r < N; r += 256) {
        float v = x[(size_t)r * F + c];
        sum += v; sq += v * v;
    }
    s_sum[t] = sum; s_sq[t] = sq;
    __syncthreads();
    for (int s = 128; s > 0; s >>= 1) {
        if (t < s) { s_sum[t] += s_sum[t + s]; s_sq[t] += s_sq[t + s]; }
        __syncthreads();
    }
    if (t == 0) {
        float m = s_sum[0] / (float)N;
        float var = s_sq[0] / (float)N - m * m;
        mu[c] = m;
        rsig[c] = rsqrtf(var + BN_EPS);
    }
}

__global__ void bn_relu_kernel(float* __restrict__ x, const float* __restrict__ mu,
                               const float* __restrict__ rsig, const float* __restrict__ g,
                               const float* __restrict__ beta, int total, int F) {
    int i = blockIdx.x * blockDim.x + threadIdx.x;
    if (i >= total) return;
    int c = i % F;
    float v = (x[i] - mu[c]) * rsig[c] * g[c] + beta[c];
    x[i] = v > 0.0f ? v : 0.0f;
}

// ---------------------------------------------------------------- WMMA GEMMs
// A 16x4 f32 fragment: lane L -> row m=L&15, K pair = 2*(L>>4){,+1}
// B 4x16 f32 fragment: lane L -> col n=L&15, K pair = 2*(L>>4){,+1}
// C/D 16x16 f32: vgpr i -> row i + 8*(L>>4), col L&15

// Layer 1: out[N,128] = [x0 @ W_init0 | x1 @ W_init1]; blockDim=256 (8 waves = 8 col tiles)
__global__ void gemm_init_wmma(const float* __restrict__ x0, const float* __restrict__ x1,
                               const float* __restrict__ Winit, float* __restrict__ out, int N) {
    const int lane = threadIdx.x & 31;
    const int colTile = threadIdx.x >> 5;            // 0..7
    const int rowBase = blockIdx.x << 4;
    const int m  = lane & 15;
    const int kh = lane >> 4;
    const int branch = colTile >> 2;
    const float* __restrict__ X = branch ? x1 : x0;
    const float* __restrict__ W = Winit + branch * (F_IN * H_DIM);
    const int nloc = ((colTile & 3) << 4) + m;       // 0..63
    int row = rowBase + m; if (row >= N) row = N - 1;
    v8f acc = {};
    for (int k = 0; k < F_IN; k += 4) {
        const int ka = k + 2 * kh;
        v2f a, b;
        a.x = X[(size_t)row * F_IN + ka];
        a.y = X[(size_t)row * F_IN + ka + 1];
        b.x = W[ka * H_DIM + nloc];
        b.y = W[(ka + 1) * H_DIM + nloc];
        acc = __builtin_amdgcn_wmma_f32_16x16x4_f32(false, a, false, b, (short)0, acc, false, false);
    }
    const int ncolG = (colTile << 4) + m;
    #pragma unroll
    for (int i = 0; i < 8; ++i) {
        int mm = rowBase + i + 8 * kh;
        if (mm < N) out[(size_t)mm * D_MID + ncolG] = acc[i];
    }
}

// Generic: out[N,ncols] = X[N,128] @ W[128,ncols]; blockDim = 32*ceil(ncols/16)
__global__ void gemm_wmma(const float* __restrict__ X, const float* __restrict__ W,
                          float* __restrict__ out, int N, int ncols) {
    const int lane = threadIdx.x & 31;
    const int colTile = threadIdx.x >> 5;
    const int rowBase = blockIdx.x << 4;
    const int m  = lane & 15;
    const int kh = lane >> 4;
    const int n = (colTile << 4) + m;
    const bool nok = n < ncols;
    int row = rowBase + m; if (row >= N) row = N - 1;
    v8f acc = {};
    for (int k = 0; k < D_MID; k += 4) {
        const int ka = k + 2 * kh;
        v2f a, b;
        a.x = X[(size_t)row * D_MID + ka];
        a.y = X[(size_t)row * D_MID + ka + 1];
        b.x = nok ? W[ka * ncols + n] : 0.0f;
        b.y = nok ? W[(ka + 1) * ncols + n] : 0.0f;
        acc = __builtin_amdgcn_wmma_f32_16x16x4_f32(false, a, false, b, (short)0, acc, false, false);
    }
    #pragma unroll
    for (int i = 0; i < 8; ++i) {
        int mm = rowBase + i + 8 * kh;
        if (mm < N && nok) out[(size_t)mm * ncols + n] = acc[i];
    }
}

// ---------------------------------------------------------------- launch
static inline int ceil_div(long a, long b) { return (int)((a + b - 1) / b); }

extern "C" void kernel_launch(void* const* d_in, const int* in_sizes, int n_in,
                              void* d_out, int out_size, void* d_ws, size_t ws_size,
                              hipStream_t stream) {
    const float* x0     = (const float*)d_in[0];
    const float* x1     = (const float*)d_in[1];
    const int*   ei     = (const int*)  d_in[2];
    const float* W_init = (const float*)d_in[3];
    // d_in[4] b_init: cancels under training-mode BN
    const float* g_init = (const float*)d_in[5];
    const float* be_init= (const float*)d_in[6];
    const float* W_mid  = (const float*)d_in[7];
    // d_in[8] b_mid: cancels under BN
    const float* g_mid  = (const float*)d_in[9];
    const float* be_mid = (const float*)d_in[10];
    const float* W_out  = (const float*)d_in[11];
    const float* b_out  = (const float*)d_in[12];
    float* out = (float*)d_out;

    const int N = in_sizes[0] / F_IN;
    const int E = in_sizes[2] / 2;
    const int* e_src = ei;
    const int* e_dst = ei + E;

    // workspace carve-out
    char* base = (char*)d_ws;
    size_t off = 0;
    auto carve = [&](size_t bytes) -> float* {
        float* p = (float*)(base + off);
        off = (off + bytes + 255) & ~(size_t)255;
        return p;
    };
    float* deg  = carve((size_t)N * 4);            // becomes dinv in place
    float* norm = carve((size_t)E * 4);
    float* hbuf = carve((size_t)N * D_MID * 4);
    float* xbuf = carve((size_t)N * D_MID * 4);
    float* mu   = carve(D_MID * 4);
    float* rsig = carve(D_MID * 4);
    (void)ws_size; (void)n_in; (void)out_size;

    const int T = 256;
    const int rowTiles = ceil_div(N, 16);

    // ---- normalization coefficients
    fill_kernel<<<ceil_div(N, T), T, 0, stream>>>(deg, 1.0f, N);                 // self-loop
    deg_kernel<<<ceil_div(E, T), T, 0, stream>>>(e_dst, deg, E);
    dinv_kernel<<<ceil_div(N, T), T, 0, stream>>>(deg, N);                       // deg -> dinv
    norm_kernel<<<ceil_div(E, T), T, 0, stream>>>(e_src, e_dst, deg, norm, E);
    const float* dinv = deg;

    // ---- layer 1: dual-branch GEMM -> hbuf[N,128]
    gemm_init_wmma<<<rowTiles, 256, 0, stream>>>(x0, x1, W_init, hbuf, N);
    // aggregate: xbuf = selfloop + edge scatter  (bias cancels in BN)
    selfloop_init_kernel<<<ceil_div((long)N * D_MID, T), T, 0, stream>>>(
        hbuf, dinv, nullptr, xbuf, N * D_MID, D_MID);
    agg_kernel<<<ceil_div((long)E * (D_MID / 4), T), T, 0, stream>>>(
        e_src, e_dst, norm, hbuf, xbuf, E, D_MID / 4, D_MID);
    bn_stats_kernel<<<D_MID, 256, 0, stream>>>(xbuf, mu, rsig, N, D_MID);
    bn_relu_kernel<<<ceil_div((long)N * D_MID, T), T, 0, stream>>>(
        xbuf, mu, rsig, g_init, be_init, N * D_MID, D_MID);

    // ---- layer 2: xbuf[N,128] @ W_mid -> hbuf[N,128]
    gemm_wmma<<<rowTiles, 256, 0, stream>>>(xbuf, W_mid, hbuf, N, D_MID);
    selfloop_init_kernel<<<ceil_div((long)N * D_MID, T), T, 0, stream>>>(
        hbuf, dinv, nullptr, xbuf, N * D_MID, D_MID);
    agg_kernel<<<ceil_div((long)E * (D_MID / 4), T), T, 0, stream>>>(
        e_src, e_dst, norm, hbuf, xbuf, E, D_MID / 4, D_MID);
    bn_stats_kernel<<<D_MID, 256, 0, stream>>>(xbuf, mu, rsig, N, D_MID);
    bn_relu_kernel<<<ceil_div((long)N * D_MID, T), T, 0, stream>>>(
        xbuf, mu, rsig, g_mid, be_mid, N * D_MID, D_MID);

    // ---- layer 3: xbuf[N,128] @ W_out[128,40] -> hbuf[N,40]; aggregate into d_out
    gemm_wmma<<<rowTiles, 96, 0, stream>>>(xbuf, W_out, hbuf, N, OUTF);
    selfloop_init_kernel<<<ceil_div((long)N * OUTF, T), T, 0, stream>>>(
        hbuf, dinv, b_out, out, N * OUTF, OUTF);
    agg_kernel<<<ceil_div((long)E * (OUTF / 4), T), T, 0, stream>>>(
        e_src, e_dst, norm, hbuf, out, E, OUTF / 4, OUTF);
}